// KPTransformer_47957604827527
// MI455X (gfx1250) — compile-verified
//
#include <hip/hip_runtime.h>
#include <math.h>

typedef __attribute__((ext_vector_type(2))) float v2f;
typedef __attribute__((ext_vector_type(8))) float v8f;

#define MQ 40000
#define NS 40000
#define HN 16
#define CC 128
#define KK 15
#define CPGc 16
#define SIGMA 2.0f
#define NEG_SLOPE 0.1f

static __device__ __forceinline__ float leaky(float x) {
    return x >= 0.0f ? x : NEG_SLOPE * x;
}

// ---------------------------------------------------------------------------
// Kernel 1: q_feats / k_feats projection GEMMs.
// One wave (32 threads) computes one 16x16 output tile with 32 chained
// V_WMMA_F32_16X16X4_F32 ops (K=128 in steps of 4). Bias folded into C init.
// grid = (N/16 row tiles, C/16 col tiles, 2 {q,k})
// ---------------------------------------------------------------------------
__global__ __launch_bounds__(32)
void qk_gemm_kernel(const float* __restrict__ s_feats,
                    const float* __restrict__ Wq, const float* __restrict__ bq,
                    const float* __restrict__ Wk, const float* __restrict__ bk,
                    float* __restrict__ qf, float* __restrict__ kf)
{
    const int lane = threadIdx.x & 31;
    const int row0 = blockIdx.x * 16;
    const int col0 = blockIdx.y * 16;
    const bool is_k = (blockIdx.z != 0);
    const float* __restrict__ W    = is_k ? Wk : Wq;
    const float* __restrict__ bias = is_k ? bk : bq;
    float* __restrict__ dst        = is_k ? kf : qf;

    const int r  = lane & 15;      // A row / B col within tile
    const int hf = lane >> 4;      // lane half selects K pair
    const int kb = hf * 2;

    v8f acc;
    const float binit = bias[col0 + r];
#pragma unroll
    for (int i = 0; i < 8; ++i) acc[i] = binit;

    const float* __restrict__ arow = s_feats + (size_t)(row0 + r) * CC;
#pragma unroll 4
    for (int k0 = 0; k0 < CC; k0 += 4) {
        v2f a, b;
        a.x = arow[k0 + kb];
        a.y = arow[k0 + kb + 1];
        b.x = W[(size_t)(k0 + kb)     * CC + col0 + r];
        b.y = W[(size_t)(k0 + kb + 1) * CC + col0 + r];
        acc = __builtin_amdgcn_wmma_f32_16x16x4_f32(
                  false, a, false, b, (short)0, acc, false, false);
    }

#pragma unroll
    for (int i = 0; i < 8; ++i)
        dst[(size_t)(row0 + hf * 8 + i) * CC + col0 + r] = acc[i];
}

// ---------------------------------------------------------------------------
// Kernel 2: fused per-query KP-Transformer.
// 256 threads = 8 waves; each wave owns one query m.
//  - geometry: nearest kernel point + influence per neighbor (VALU)
//  - MLP1: leaky(q-k) (16x128) @ alpha_w1 (128x16)  -> 32 WMMAs
//  - MLP2: leaky(.)   (16x16)  @ alpha_w2 (16x16)+b -> 4 WMMAs (via LDS relayout)
//  - sigmoid -> LDS, then float4 weighted aggregation over 16 neighbors
// ---------------------------------------------------------------------------
__global__ __launch_bounds__(256)
void kpt_fused_kernel(const float* __restrict__ q_pts,
                      const float* __restrict__ s_pts,
                      const float* __restrict__ s_feats,
                      const int*   __restrict__ neighb_inds,
                      const float* __restrict__ kernel_points,
                      const float* __restrict__ weights,
                      const float* __restrict__ w1,
                      const float* __restrict__ w2,
                      const float* __restrict__ b2,
                      const float* __restrict__ qf,
                      const float* __restrict__ kf,
                      float* __restrict__ out)
{
    const int lane = threadIdx.x & 31;
    const int wid  = threadIdx.x >> 5;
    const int m    = blockIdx.x * 8 + wid;

    __shared__ float s_a  [8][256];   // 16x16 MLP1 output per wave
    __shared__ float s_att[8][256];   // 16x16 attention per wave
    __shared__ int   s_idx [8][16];
    __shared__ int   s_nn  [8][16];
    __shared__ float s_infl[8][16];

    const int h  = lane & 15;         // neighbor / matrix row / tile col
    const int hf = lane >> 4;
    const int kb = hf * 2;

    // ---- geometry: nearest kernel point + influence (both halves duplicate) ----
    {
        const int idx = neighb_inds[(size_t)m * HN + h];
        const float qx = q_pts[(size_t)m * 3 + 0];
        const float qy = q_pts[(size_t)m * 3 + 1];
        const float qz = q_pts[(size_t)m * 3 + 2];
        const float dx = s_pts[(size_t)idx * 3 + 0] - qx;
        const float dy = s_pts[(size_t)idx * 3 + 1] - qy;
        const float dz = s_pts[(size_t)idx * 3 + 2] - qz;
        float best = 3.0e38f;
        int   bi   = 0;
#pragma unroll
        for (int k = 0; k < KK; ++k) {
            const float ex = dx - kernel_points[k * 3 + 0];
            const float ey = dy - kernel_points[k * 3 + 1];
            const float ez = dz - kernel_points[k * 3 + 2];
            const float sq = ex * ex + ey * ey + ez * ez;
            if (sq < best) { best = sq; bi = k; }
        }
        float infl = 1.0f - sqrtf(best) * (1.0f / SIGMA);
        infl = infl > 0.0f ? infl : 0.0f;
        if (lane < 16) {
            s_idx [wid][h] = idx;
            s_nn  [wid][h] = bi;
            s_infl[wid][h] = infl;
        }
    }
    __syncthreads();

    const int nbr = s_idx[wid][h];

    // ---- MLP1: A[h,k] = leaky(qf[m,k] - kf[nbr,k]), B = alpha_w1 (128x16) ----
    v8f acc;
#pragma unroll
    for (int i = 0; i < 8; ++i) acc[i] = 0.0f;

    const float* __restrict__ qrow = qf + (size_t)m   * CC;
    const float* __restrict__ krow = kf + (size_t)nbr * CC;
#pragma unroll 4
    for (int k0 = 0; k0 < CC; k0 += 4) {
        v2f a, b;
        a.x = leaky(qrow[k0 + kb]     - krow[k0 + kb]);
        a.y = leaky(qrow[k0 + kb + 1] - krow[k0 + kb + 1]);
        b.x = w1[(k0 + kb)     * CPGc + h];
        b.y = w1[(k0 + kb + 1) * CPGc + h];
        acc = __builtin_amdgcn_wmma_f32_16x16x4_f32(
                  false, a, false, b, (short)0, acc, false, false);
    }

    // leaky + C-layout -> LDS (row-major 16x16)
#pragma unroll
    for (int i = 0; i < 8; ++i)
        s_a[wid][(hf * 8 + i) * 16 + h] = leaky(acc[i]);
    __syncthreads();

    // ---- MLP2: (16x16) @ alpha_w2 + b2 ----
    v8f acc2;
    const float bb = b2[h];
#pragma unroll
    for (int i = 0; i < 8; ++i) acc2[i] = bb;

#pragma unroll
    for (int k0 = 0; k0 < 16; k0 += 4) {
        v2f a, b;
        a.x = s_a[wid][h * 16 + k0 + kb];
        a.y = s_a[wid][h * 16 + k0 + kb + 1];
        b.x = w2[(k0 + kb)     * CPGc + h];
        b.y = w2[(k0 + kb + 1) * CPGc + h];
        acc2 = __builtin_amdgcn_wmma_f32_16x16x4_f32(
                   false, a, false, b, (short)0, acc2, false, false);
    }

    // sigmoid -> LDS attention (row = neighbor h, col = channel-in-group j)
#pragma unroll
    for (int i = 0; i < 8; ++i)
        s_att[wid][(hf * 8 + i) * 16 + h] = 1.0f / (1.0f + __expf(-acc2[i]));
    __syncthreads();

    // ---- aggregation: out[m,c] = sum_h v[h,c]*w[nn(h),c]*infl(h)*att[h,c&15] ----
    const int c0 = lane * 4;          // 4 consecutive channels per lane
    const int j0 = c0 & 15;
    float o0 = 0.0f, o1 = 0.0f, o2 = 0.0f, o3 = 0.0f;
#pragma unroll
    for (int hh = 0; hh < HN; ++hh) {
        const int   idx = s_idx [wid][hh];
        const int   nn  = s_nn  [wid][hh];
        const float fl  = s_infl[wid][hh];
        const float4 v = *(const float4*)(s_feats + (size_t)idx * CC + c0);
        const float4 w = *(const float4*)(weights + (size_t)nn  * CC + c0);
        const float4 a = *(const float4*)(&s_att[wid][hh * 16 + j0]);
        o0 += v.x * (w.x * fl) * a.x;
        o1 += v.y * (w.y * fl) * a.y;
        o2 += v.z * (w.z * fl) * a.z;
        o3 += v.w * (w.w * fl) * a.w;
    }
    float4 res = make_float4(o0, o1, o2, o3);
    *(float4*)(out + (size_t)m * CC + c0) = res;
}

// ---------------------------------------------------------------------------
extern "C" void kernel_launch(void* const* d_in, const int* in_sizes, int n_in,
                              void* d_out, int out_size, void* d_ws, size_t ws_size,
                              hipStream_t stream) {
    (void)in_sizes; (void)n_in; (void)out_size; (void)ws_size;
    const float* q_pts         = (const float*)d_in[0];
    const float* s_pts         = (const float*)d_in[1];
    const float* s_feats       = (const float*)d_in[2];
    const int*   neighb_inds   = (const int*)  d_in[3];
    const float* kernel_points = (const float*)d_in[4];
    const float* weights       = (const float*)d_in[5];
    const float* Wq            = (const float*)d_in[6];
    const float* bq            = (const float*)d_in[7];
    const float* Wk            = (const float*)d_in[8];
    const float* bk            = (const float*)d_in[9];
    const float* w1            = (const float*)d_in[10];
    const float* w2            = (const float*)d_in[11];
    const float* b2            = (const float*)d_in[12];
    float* out = (float*)d_out;

    float* qf = (float*)d_ws;                     // (N, C)
    float* kf = qf + (size_t)NS * CC;             // (N, C)

    dim3 g1(NS / 16, CC / 16, 2);
    qk_gemm_kernel<<<g1, 32, 0, stream>>>(s_feats, Wq, bq, Wk, bk, qf, kf);

    kpt_fused_kernel<<<MQ / 8, 256, 0, stream>>>(
        q_pts, s_pts, s_feats, neighb_inds, kernel_points, weights,
        w1, w2, b2, qf, kf, out);
}